// MS_SSIM_90941637526260
// MI455X (gfx1250) — compile-verified
//
#include <hip/hip_runtime.h>
#include <math.h>

// ============================================================================
// MS-SSIM + L1(Gauss) + color loss for MI455X (gfx1250, wave32, WMMA).
//
// Design (compile-only reasoning, see analysis):
//  * inputs (151 MB) fit in 192 MB L2 -> fully fused tile kernels, halo
//    re-reads hit L2, not HBM.
//  * all Gaussian windows are separable -> 1D h-pass + 1D v-pass.
//  * horizontal 1D conv == banded GEMM -> V_WMMA_F32_16X16X4_F32 (f32 in/out,
//    matches f32 reference precision, engages CDNA5 matrix pipes).
//  * only persistent scratch: image pyramid (~33.4 MB) + 128 accumulator
//    floats at the head of d_ws.
// ============================================================================

typedef __attribute__((ext_vector_type(2))) float v2f;
typedef __attribute__((ext_vector_type(8))) float v8f;

#define DEVFN static __device__ __forceinline__

#define C1_SSIM 6.5025f    // (0.01*255)^2
#define C2_SSIM 58.5225f   // (0.03*255)^2
#define HB_STR  17         // h-buffer LDS row stride (odd -> bank friendly)

// accumulator layout (floats, at d_ws[0]):
//  [0..4] ssim_sum[level]  [5..9] mcs_sum[level]  [10] l1_sum  [11] clr_sum
//  [12..27] me1[n]  [28..43] me3[n]
//  [48..58] w11   [59..65] w7   [66..94] w29   (Gaussian vectors)
// pyramid buffers start at float offset 128.

DEVFN void lds_fence() { asm volatile("s_wait_dscnt 0" ::: "memory"); }

DEVFN float wave_sum(float v) {
#pragma unroll
  for (int off = 16; off > 0; off >>= 1) v += __shfl_xor(v, off, 32);
  return v;
}

DEVFN float poly5(float x) {
  // cw[5]*x^5 + ... + cw[0], Horner
  return ((((1.556091f * x - 1.894404f) * x + 1.435936f) * x - 0.173433f) * x +
          0.07633f) * x - 0.000519f;
}

DEVFN bool tile_coords(int gtile, int planes, int H, int W,
                       int& plane, int& y0, int& x0) {
  const int tx = W >> 4, ty = H >> 4, tpp = tx * ty;
  if (gtile >= planes * tpp) return false;
  plane = gtile / tpp;
  int r = gtile - plane * tpp;
  int ry = r / tx;
  y0 = ry * 16;
  x0 = (r - ry * tx) * 16;
  return true;
}

// ---- staging: tile + halo into LDS with zero padding -----------------------
template <int SR, int SC, int SSTR>
DEVFN void stage1(float* dst, const float* __restrict__ src, int H, int W,
                  int y0, int x0, int pad, int lane) {
  for (int i = lane; i < SR * SC; i += 32) {
    int r = i / SC, c = i - r * SC;
    int y = y0 - pad + r, x = x0 - pad + c;
    float v = 0.f;
    if ((unsigned)y < (unsigned)H && (unsigned)x < (unsigned)W)
      v = src[(size_t)y * W + x];
    dst[r * SSTR + c] = v;
  }
}

template <int SR, int SC, int SSTR>
DEVFN void stage_absdiff(float* dst, const float* __restrict__ a,
                         const float* __restrict__ b, int H, int W, int y0,
                         int x0, int pad, int lane) {
  for (int i = lane; i < SR * SC; i += 32) {
    int r = i / SC, c = i - r * SC;
    int y = y0 - pad + r, x = x0 - pad + c;
    float v = 0.f;
    if ((unsigned)y < (unsigned)H && (unsigned)x < (unsigned)W) {
      size_t o = (size_t)y * W + x;
      v = fabsf(a[o] - b[o]);
    }
    dst[r * SSTR + c] = v;
  }
}

// ---- banded B operand for the horizontal conv GEMM -------------------------
// B[k][n] = w[k-n] (0 <= k-n < T).  Lane l holds B[(l>>4)*2 + {0,1}][l&15].
template <int T, int CH>
DEVFN void make_b(v2f* b, const float* __restrict__ wg, int lane) {
  const int n = lane & 15, hi = lane >> 4;
#pragma unroll
  for (int ch = 0; ch < CH; ++ch) {
    int k0 = ch * 4 + hi * 2;
    int t0 = k0 - n, t1 = t0 + 1;
    v2f bb;
    bb[0] = (t0 >= 0 && t0 < T) ? wg[t0] : 0.f;
    bb[1] = (t1 >= 0 && t1 < T) ? wg[t1] : 0.f;
    b[ch] = bb;
  }
}

// source transform at A-operand fetch time (SSIM needs x1,x2,x1^2,x2^2,x1x2)
template <int MODE>
DEVFN float fetchv(const float* s1, const float* s2, int idx) {
  if (MODE == 0) return s1[idx];
  if (MODE == 1) return s2[idx];
  if (MODE == 2) { float a = s1[idx]; return a * a; }
  if (MODE == 3) { float a = s2[idx]; return a * a; }
  return s1[idx] * s2[idx];
}

// ---- horizontal 1D conv via V_WMMA_F32_16X16X4_F32 -------------------------
// hbuf[r][n] = sum_k stage[r][k] * w[k-n], for r = 0 .. RT*16-1
template <int T, int CH, int RT, int SSTR, int MODE>
DEVFN void hconv(const float* s1, const float* s2, const v2f* bv, float* hb,
                 int lane) {
  const int m = lane & 15, hi = lane >> 4;
#pragma unroll
  for (int rt = 0; rt < RT; ++rt) {
    v8f acc = {0.f, 0.f, 0.f, 0.f, 0.f, 0.f, 0.f, 0.f};
#pragma unroll
    for (int ch = 0; ch < CH; ++ch) {
      const int base = (rt * 16 + m) * SSTR + ch * 4 + hi * 2;
      v2f a;
      a[0] = fetchv<MODE>(s1, s2, base);
      a[1] = fetchv<MODE>(s1, s2, base + 1);
      acc = __builtin_amdgcn_wmma_f32_16x16x4_f32(
          false, a, false, bv[ch], (short)0, acc, false, false);
    }
#pragma unroll
    for (int v = 0; v < 8; ++v)
      hb[(rt * 16 + v + 8 * hi) * HB_STR + m] = acc[v];
  }
}

// ---- vertical 1D conv (VALU), 8 output pixels per lane ---------------------
template <int T>
DEVFN void vconv(const float* hb, const float* w, float out[8], int lane) {
  const int m15 = lane & 15, hi = lane >> 4;
#pragma unroll
  for (int v = 0; v < 8; ++v) {
    const int m = v + 8 * hi;
    float s = 0.f;
#pragma unroll
    for (int t = 0; t < T; ++t) s += w[t] * hb[(m + t) * HB_STR + m15];
    out[v] = s;
  }
}

// ============================================================================
// Kernels
// ============================================================================

__global__ void init_kernel(float* acc) {
  int t = threadIdx.x;
  if (t < 48) acc[t] = 0.f;
  if (t == 0) {
    // Gaussian vectors (normalized 1D; 2D window is the exact outer product)
    const int   ws[3] = {11, 7, 29};
    const float sg[3] = {1.5f, 1.1f, 29.f / 6.f};
    const int   of[3] = {48, 59, 66};
    for (int j = 0; j < 3; ++j) {
      float s = 0.f;
      for (int i = 0; i < ws[j]; ++i) {
        float d = (float)(i - ws[j] / 2);
        float g = expf(-(d * d) / (2.f * sg[j] * sg[j]));
        acc[of[j] + i] = g;
        s += g;
      }
      for (int i = 0; i < ws[j]; ++i) acc[of[j] + i] /= s;
    }
  }
}

// ---- SSIM level: 5 fused separable convs + map + reduction -----------------
__global__ __launch_bounds__(128) void ssim_kernel(
    const float* __restrict__ img1, const float* __restrict__ img2, int H,
    int W, int planes, float* __restrict__ acc, int level) {
  constexpr int T = 11, CH = 7, RT = 2, SC = 28, SSTR = 30, SR = 32;
  static_assert(4 * CH >= 15 + T, "band coverage");
  __shared__ float s1m[4][SR * SSTR], s2m[4][SR * SSTR],
      hbm[4][RT * 16 * HB_STR];
  const int wid = threadIdx.x >> 5, lane = threadIdx.x & 31;
  int plane, y0, x0;
  if (!tile_coords(blockIdx.x * 4 + wid, planes, H, W, plane, y0, x0)) return;
  float* s1 = s1m[wid];
  float* s2 = s2m[wid];
  float* hb = hbm[wid];
  const float* p1 = img1 + (size_t)plane * H * W;
  const float* p2 = img2 + (size_t)plane * H * W;
  __builtin_prefetch(p1 + (size_t)y0 * W + x0, 0, 1);   // global_prefetch_b8
  __builtin_prefetch(p2 + (size_t)y0 * W + x0, 0, 1);
  stage1<SR, SC, SSTR>(s1, p1, H, W, y0, x0, 5, lane);
  stage1<SR, SC, SSTR>(s2, p2, H, W, y0, x0, 5, lane);
  lds_fence();
  const float* wg = acc + 48;
  float w[T];
#pragma unroll
  for (int t = 0; t < T; ++t) w[t] = wg[t];
  v2f bv[CH];
  make_b<T, CH>(bv, wg, lane);

  float mu1[8], mu2[8], s11[8], s22[8], s12[8];
  hconv<T, CH, RT, SSTR, 0>(s1, s2, bv, hb, lane); lds_fence();
  vconv<T>(hb, w, mu1, lane);                       lds_fence();
  hconv<T, CH, RT, SSTR, 1>(s1, s2, bv, hb, lane); lds_fence();
  vconv<T>(hb, w, mu2, lane);                       lds_fence();
  hconv<T, CH, RT, SSTR, 2>(s1, s2, bv, hb, lane); lds_fence();
  vconv<T>(hb, w, s11, lane);                       lds_fence();
  hconv<T, CH, RT, SSTR, 3>(s1, s2, bv, hb, lane); lds_fence();
  vconv<T>(hb, w, s22, lane);                       lds_fence();
  hconv<T, CH, RT, SSTR, 4>(s1, s2, bv, hb, lane); lds_fence();
  vconv<T>(hb, w, s12, lane);

  float ssum = 0.f, msum = 0.f;
#pragma unroll
  for (int v = 0; v < 8; ++v) {
    float m1 = mu1[v], m2 = mu2[v];
    float sg1 = s11[v] - m1 * m1;
    float sg2 = s22[v] - m2 * m2;
    float sgx = s12[v] - m1 * m2;
    float V1 = 2.f * sgx + C2_SSIM;
    float V2 = sg1 + sg2 + C2_SSIM;
    ssum += ((2.f * m1 * m2 + C1_SSIM) * V1) /
            ((m1 * m1 + m2 * m2 + C1_SSIM) * V2);
    msum += V1 / V2;
  }
  ssum = wave_sum(ssum);
  msum = wave_sum(msum);
  if (lane == 0) {
    atomicAdd(&acc[level], ssum);
    atomicAdd(&acc[5 + level], msum);
  }
}

// ---- L1 through 7x7 Gaussian -----------------------------------------------
__global__ __launch_bounds__(128) void l1_kernel(
    const float* __restrict__ img1, const float* __restrict__ img2, int H,
    int W, int planes, float* __restrict__ acc) {
  constexpr int T = 7, CH = 6, RT = 2, SC = 24, SSTR = 26, SR = 32;
  static_assert(4 * CH >= 15 + T, "band coverage");
  __shared__ float sm[4][SR * SSTR], hbm[4][RT * 16 * HB_STR];
  const int wid = threadIdx.x >> 5, lane = threadIdx.x & 31;
  int plane, y0, x0;
  if (!tile_coords(blockIdx.x * 4 + wid, planes, H, W, plane, y0, x0)) return;
  float* s = sm[wid];
  float* hb = hbm[wid];
  const float* p1 = img1 + (size_t)plane * H * W;
  const float* p2 = img2 + (size_t)plane * H * W;
  stage_absdiff<SR, SC, SSTR>(s, p1, p2, H, W, y0, x0, 3, lane);
  lds_fence();
  const float* wg = acc + 59;
  float w[T];
#pragma unroll
  for (int t = 0; t < T; ++t) w[t] = wg[t];
  v2f bv[CH];
  make_b<T, CH>(bv, wg, lane);
  float r[8];
  hconv<T, CH, RT, SSTR, 0>(s, s, bv, hb, lane);
  lds_fence();
  vconv<T>(hb, w, r, lane);
  float sum = r[0] + r[1] + r[2] + r[3] + r[4] + r[5] + r[6] + r[7];
  sum = wave_sum(sum);
  if (lane == 0) atomicAdd(&acc[10], sum);
}

// ---- color pass 1: per-image gray sums me1/me3 -----------------------------
__global__ __launch_bounds__(128) void clr_pass1_kernel(
    const float* __restrict__ img, int H, int W, int planes,
    float* __restrict__ acc, int apply_poly, int accbase) {
  constexpr int T = 29, CH = 11, RT = 3, SC = 44, SSTR = 46, SR = 48;
  static_assert(4 * CH >= 15 + T, "band coverage");
  __shared__ float sm[4][SR * SSTR], hbm[4][RT * 16 * HB_STR];
  const int wid = threadIdx.x >> 5, lane = threadIdx.x & 31;
  int plane, y0, x0;
  if (!tile_coords(blockIdx.x * 4 + wid, planes, H, W, plane, y0, x0)) return;
  float* s = sm[wid];
  float* hb = hbm[wid];
  const float* p = img + (size_t)plane * H * W;
  stage1<SR, SC, SSTR>(s, p, H, W, y0, x0, 14, lane);
  lds_fence();
  const float* wg = acc + 66;
  float w[T];
#pragma unroll
  for (int t = 0; t < T; ++t) w[t] = wg[t];
  v2f bv[CH];
  make_b<T, CH>(bv, wg, lane);
  float mu[8];
  hconv<T, CH, RT, SSTR, 0>(s, s, bv, hb, lane);
  lds_fence();
  vconv<T>(hb, w, mu, lane);
  const int c = plane % 3, n = plane / 3;
  const float gw = (c == 0) ? 0.299f : ((c == 1) ? 0.587f : 0.144f);
  float sum = 0.f;
#pragma unroll
  for (int v = 0; v < 8; ++v) {
    float m = mu[v];
    if (apply_poly) m = poly5(m);
    sum += gw * m;
  }
  sum = wave_sum(sum);
  if (lane == 0) atomicAdd(&acc[accbase + n], sum);
}

// ---- color pass 2: |poly(mu1) - mu3*sc| (recompute convs; L2 holds inputs) -
__global__ __launch_bounds__(128) void clr_pass2_kernel(
    const float* __restrict__ img1, const float* __restrict__ img3, int H,
    int W, int planes, float* __restrict__ acc) {
  constexpr int T = 29, CH = 11, RT = 3, SC = 44, SSTR = 46, SR = 48;
  __shared__ float sm[4][SR * SSTR], hbm[4][RT * 16 * HB_STR];
  const int wid = threadIdx.x >> 5, lane = threadIdx.x & 31;
  int plane, y0, x0;
  if (!tile_coords(blockIdx.x * 4 + wid, planes, H, W, plane, y0, x0)) return;
  float* s = sm[wid];
  float* hb = hbm[wid];
  const float* wg = acc + 66;
  float w[T];
#pragma unroll
  for (int t = 0; t < T; ++t) w[t] = wg[t];
  v2f bv[CH];
  make_b<T, CH>(bv, wg, lane);

  float mu1p[8], mu3[8];
  stage1<SR, SC, SSTR>(s, img1 + (size_t)plane * H * W, H, W, y0, x0, 14, lane);
  lds_fence();
  hconv<T, CH, RT, SSTR, 0>(s, s, bv, hb, lane);
  lds_fence();
  vconv<T>(hb, w, mu1p, lane);
  lds_fence();  // WAR: re-stage same LDS with img3
  stage1<SR, SC, SSTR>(s, img3 + (size_t)plane * H * W, H, W, y0, x0, 14, lane);
  lds_fence();
  hconv<T, CH, RT, SSTR, 0>(s, s, bv, hb, lane);
  lds_fence();
  vconv<T>(hb, w, mu3, lane);

  const int n = plane / 3;
  const float sc = acc[12 + n] / acc[28 + n];  // (H*W) factors cancel
  float sum = 0.f;
#pragma unroll
  for (int v = 0; v < 8; ++v) sum += fabsf(poly5(mu1p[v]) - mu3[v] * sc);
  sum = wave_sum(sum);
  if (lane == 0) atomicAdd(&acc[11], sum);
}

// ---- 2x2 average pool ------------------------------------------------------
__global__ void downsample_kernel(const float* __restrict__ src,
                                  float* __restrict__ dst, int H, int W,
                                  int planes) {
  const int H2 = H >> 1, W2 = W >> 1;
  const size_t total = (size_t)planes * H2 * W2;
  for (size_t i = (size_t)blockIdx.x * blockDim.x + threadIdx.x; i < total;
       i += (size_t)gridDim.x * blockDim.x) {
    int x = (int)(i % W2);
    size_t r = i / W2;
    int y = (int)(r % H2);
    int p = (int)(r / H2);
    const float* sp = src + ((size_t)p * H + 2 * y) * W + 2 * x;
    dst[i] = 0.25f * (sp[0] + sp[1] + sp[W] + sp[W + 1]);
  }
}

// ---- combine everything into the scalar loss -------------------------------
__global__ void finalize_kernel(const float* __restrict__ acc,
                                float* __restrict__ out) {
  const float wts[5] = {0.0448f, 0.2856f, 0.3001f, 0.2363f, 0.1333f};
  const float P0 = 16.f * 3.f * 512.f * 512.f;
  float value = 1.f;
  for (int L = 0; L < 5; ++L) {
    float cnt = 16.f * 3.f * (float)(512 >> L) * (float)(512 >> L);
    float sm = acc[L] / cnt;
    float mm = acc[5 + L] / cnt;
    value *= (L < 4) ? powf(mm, wts[L]) : powf(sm, wts[4]);
  }
  const float ssimv = 1.f - value;
  const float l1v = acc[10] / P0;
  const float clrv = acc[11] / P0;
  out[0] = 0.4f * ssimv + 0.4f * l1v + 0.2f * clrv;
}

// ============================================================================
extern "C" void kernel_launch(void* const* d_in, const int* in_sizes, int n_in,
                              void* d_out, int out_size, void* d_ws,
                              size_t ws_size, hipStream_t stream) {
  const float* img1 = (const float*)d_in[0];
  const float* img2 = (const float*)d_in[1];
  const float* img3 = (const float*)d_in[2];
  float* acc = (float*)d_ws;  // 128 floats of accumulators + weights
  // pyramid buffers (levels 1..4, both images): ~33.4 MB of d_ws
  float* buf = acc + 128;
  float* q1[4];
  float* q2[4];
  size_t off = 0;
  for (int L = 1; L <= 4; ++L) {
    size_t s = 48ull * (size_t)(512 >> L) * (size_t)(512 >> L);
    q1[L - 1] = buf + off; off += s;
    q2[L - 1] = buf + off; off += s;
  }

  init_kernel<<<1, 64, 0, stream>>>(acc);

  const int planes = 48;  // 16 * 3
  auto nblocks = [](int H, int W) {
    int tiles = 48 * (H / 16) * (W / 16);
    return (tiles + 3) / 4;  // 4 waves / block, one 16x16 tile / wave
  };

  const int b0 = nblocks(512, 512);
  l1_kernel<<<b0, 128, 0, stream>>>(img1, img2, 512, 512, planes, acc);
  clr_pass1_kernel<<<b0, 128, 0, stream>>>(img1, 512, 512, planes, acc, 1, 12);
  clr_pass1_kernel<<<b0, 128, 0, stream>>>(img3, 512, 512, planes, acc, 0, 28);
  clr_pass2_kernel<<<b0, 128, 0, stream>>>(img1, img3, 512, 512, planes, acc);

  const float* a = img1;
  const float* b = img2;
  for (int L = 0; L < 5; ++L) {
    const int H = 512 >> L;
    ssim_kernel<<<nblocks(H, H), 128, 0, stream>>>(a, b, H, H, planes, acc, L);
    if (L < 4) {
      size_t tot = 48ull * (size_t)(H / 2) * (size_t)(H / 2);
      int blocks = (int)((tot + 255) / 256);
      downsample_kernel<<<blocks, 256, 0, stream>>>(a, q1[L], H, H, planes);
      downsample_kernel<<<blocks, 256, 0, stream>>>(b, q2[L], H, H, planes);
      a = q1[L];
      b = q2[L];
    }
  }

  finalize_kernel<<<1, 1, 0, stream>>>(acc, (float*)d_out);
}